// GroupRouter_17428977287675
// MI455X (gfx1250) — compile-verified
//
#include <hip/hip_runtime.h>
#include <hip/hip_bf16.h>
#include <math.h>

typedef __attribute__((ext_vector_type(2))) float v2f;
typedef __attribute__((ext_vector_type(8))) float v8f;

#define LN_EPS      1e-5f
#define RENORM_EPS  1e-8f
#define ANTI_W      0.1f

#define D_MODEL 2048
#define N_EXP   64
#define KC      128
#define NCHUNK  (D_MODEL / KC)
#define XSTRIDE 132            // padded x-tile row stride (floats): 16B-aligned rows, 4-bank skew

// ---- CDNA5 async global->LDS copy (ASYNCcnt-tracked, ISA 15.18.3 op 96/98) ----
__device__ __forceinline__ void async_b128(uint32_t lds_off, const float* g) {
    asm volatile("global_load_async_to_lds_b128 %0, %1, off"
                 :: "v"(lds_off), "v"((unsigned long long)(uintptr_t)g) : "memory");
}
__device__ __forceinline__ void async_b32(uint32_t lds_off, const float* g) {
    asm volatile("global_load_async_to_lds_b32 %0, %1, off"
                 :: "v"(lds_off), "v"((unsigned long long)(uintptr_t)g) : "memory");
}
#define WAIT_ASYNC(n) asm volatile("s_wait_asynccnt " #n ::: "memory")

// ---- precompute gdot[n] = gamma.W[n], bdot[n] = beta.W[n] into ws[64..191] ----
__global__ __launch_bounds__(256) void precompute_dots(
    const float* __restrict__ W, const float* __restrict__ gamma,
    const float* __restrict__ beta, float* __restrict__ ws)
{
    __shared__ float rg[256], rb[256];
    const int n = blockIdx.x;
    const float* wr = W + (long)n * D_MODEL;
    float g = 0.f, b = 0.f;
    for (int k = threadIdx.x; k < D_MODEL; k += 256) {
        float w = wr[k];
        g += gamma[k] * w;
        b += beta[k] * w;
    }
    rg[threadIdx.x] = g; rb[threadIdx.x] = b;
    __syncthreads();
    for (int s2 = 128; s2 > 0; s2 >>= 1) {
        if (threadIdx.x < s2) {
            rg[threadIdx.x] += rg[threadIdx.x + s2];
            rb[threadIdx.x] += rb[threadIdx.x + s2];
        }
        __syncthreads();
    }
    if (threadIdx.x == 0) { ws[N_EXP + n] = rg[0]; ws[2 * N_EXP + n] = rb[0]; }
}

__global__ __launch_bounds__(128) void router_main(
    const float* __restrict__ x, const float* __restrict__ W,
    const float* __restrict__ bias, const float* __restrict__ gamma,
    float* __restrict__ out_sparse, float* __restrict__ out_idx,
    float* __restrict__ ws)   // [0:64) acc, [64:128) gdot, [128:192) bdot
{
    __shared__ float sX[2][16 * XSTRIDE];   // raw x tiles (double-buffered)
    __shared__ float sG[2][KC];             // gamma chunks
    __shared__ float s_mu[16], s_rs[16];
    __shared__ float s_logit[16][N_EXP];
    __shared__ float s_acc[N_EXP];
    __shared__ float s_v1[16], s_v2[16];
    __shared__ int   s_i1[16], s_i2[16];

    const int  t    = threadIdx.x;
    const int  wave = t >> 5;
    const int  lane = t & 31;
    const long rowBase = (long)blockIdx.x * 16;

    if (t < N_EXP) s_acc[t] = 0.f;

    // per-chunk async issue: 4 x B128 (x tile) + 1 x B32 (gamma) per thread
    auto issue = [&](int ci, int buf) {
        const int kc = ci * KC;
#pragma unroll
        for (int j = 0; j < 4; ++j) {
            const int f   = (t + 128 * j) * 4;       // flat float idx in 16xKC tile
            const int row = f >> 7;
            const int k   = f & (KC - 1);
            async_b128((uint32_t)(uintptr_t)&sX[buf][row * XSTRIDE + k],
                       x + (rowBase + row) * D_MODEL + kc + k);
        }
        async_b32((uint32_t)(uintptr_t)&sG[buf][t], gamma + kc + t);
    };

    const int nl  = lane & 15;         // A row / B column within tile
    const int kh  = lane >> 4;         // K-pair selector
    const int myN = wave * 16 + nl;
    const float* wrow = W + (long)myN * D_MODEL;

    const int srow = t >> 3;           // stats: 8 threads per row
    const int ssub = t & 7;
    float st_s = 0.f, st_ss = 0.f;

    v8f c = {};
    issue(0, 0);

    for (int ci = 0; ci < NCHUNK; ++ci) {
        const int cur = ci & 1;
        if (ci + 1 < NCHUNK) {
            issue(ci + 1, cur ^ 1);
            WAIT_ASYNC(5);             // next chunk's 5 ops may remain in flight
        } else {
            WAIT_ASYNC(0);
        }
        __syncthreads();

        // LN statistics from the staged raw-x tile (single global read of x)
#pragma unroll 4
        for (int i = 0; i < KC / 8; ++i) {
            float v = sX[cur][srow * XSTRIDE + ssub + 8 * i];
            st_s += v; st_ss += v * v;
        }

        // GEMM: A = x (gamma applied at read), B = W streamed from L2
        const float* wk = wrow + ci * KC;
#pragma unroll 4
        for (int k0 = 0; k0 < KC; k0 += 4) {
            const int ka = k0 + 2 * kh;
            v2f g = *(const v2f*)&sG[cur][ka];
            v2f a = *(const v2f*)&sX[cur][nl * XSTRIDE + ka];
            a = a * g;
            v2f b = *(const v2f*)(wk + ka);
            c = __builtin_amdgcn_wmma_f32_16x16x4_f32(
                    false, a, false, b, (short)0, c, false, false);
        }
        __syncthreads();               // everyone done with buf `cur` before reuse
    }

    // ---- finish LN stats ----
    for (int m = 4; m >= 1; m >>= 1) {
        st_s  += __shfl_xor(st_s,  m, 32);
        st_ss += __shfl_xor(st_ss, m, 32);
    }
    if (ssub == 0) {
        float mu  = st_s / (float)D_MODEL;
        float var = st_ss / (float)D_MODEL - mu * mu;
        s_mu[srow] = mu;
        s_rs[srow] = rsqrtf(var + LN_EPS);
    }
    __syncthreads();

    // ---- logits: fold LN back in: rs*cdot - rs*mu*gdot + bdot + bias ----
    {
        const int rb2 = kh * 8;
        const float gd = ws[N_EXP + myN];
        const float bd = ws[2 * N_EXP + myN];
        const float bn = bias[myN];
#pragma unroll
        for (int g = 0; g < 8; ++g) {
            const int row = rb2 + g;
            const float rs = s_rs[row];
            s_logit[row][myN] = rs * c[g] - rs * s_mu[row] * gd + bd + bn;
        }
    }
    __syncthreads();

    // ---- per-row softmax + top-2 (threads 0..15) ----
    if (t < 16) {
        const float* l = s_logit[t];
        float m = l[0];
        for (int n = 1; n < N_EXP; ++n) m = fmaxf(m, l[n]);
        float ssum = 0.f;
        for (int n = 0; n < N_EXP; ++n) ssum += __expf(l[n] - m);
        const float inv = 1.f / ssum;
        float v1 = -1.f, v2 = -1.f; int i1 = 0, i2 = 0;
        for (int n = 0; n < N_EXP; ++n) {
            float w = __expf(l[n] - m) * inv;
            atomicAdd(&s_acc[n], w);
            if (w > v1)      { v2 = v1; i2 = i1; v1 = w; i1 = n; }
            else if (w > v2) { v2 = w;  i2 = n; }
        }
        const float rn = 1.f / (v1 + v2 + RENORM_EPS);
        s_i1[t] = i1; s_i2[t] = i2;
        s_v1[t] = v1 * rn; s_v2[t] = v2 * rn;
        const long gr = rowBase + t;
        out_idx[gr * 2 + 0] = (float)i1;
        out_idx[gr * 2 + 1] = (float)i2;
    }
    __syncthreads();

    // ---- coalesced sparse-weight write ----
    for (int e = t; e < 16 * N_EXP; e += 128) {
        int row = e >> 6;
        int n   = e & (N_EXP - 1);
        float v = 0.f;
        if      (n == s_i1[row]) v = s_v1[row];
        else if (n == s_i2[row]) v = s_v2[row];
        out_sparse[(rowBase + row) * N_EXP + n] = v;
    }
    if (t < N_EXP) atomicAdd(&ws[t], s_acc[t]);
}

__global__ __launch_bounds__(64) void router_finalize(
    const float* __restrict__ acc, const float* __restrict__ ema,
    float* __restrict__ out_loss, int rows, int nexp)
{
    __shared__ float s_t[N_EXP], s_e[N_EXP];
    const int t = threadIdx.x;
    float mean = acc[t] / (float)rows;
    s_t[t] = mean * __logf(mean + RENORM_EPS);
    s_e[t] = ema[t];
    __syncthreads();
    if (t == 0) {
        float s = 0.f, mx = -1e30f;
        for (int n = 0; n < nexp; ++n) { s += s_t[n]; mx = fmaxf(mx, s_e[n]); }
        const float uniform = 1.f / (float)nexp;
        const float margin  = fminf(0.15f, (1.f - uniform) * 0.3f);
        const float thr     = uniform + margin;
        const float pen     = fmaxf(0.f, mx - thr);
        *out_loss = s + ANTI_W * pen;
    }
}

extern "C" void kernel_launch(void* const* d_in, const int* in_sizes, int n_in,
                              void* d_out, int out_size, void* d_ws, size_t ws_size,
                              hipStream_t stream) {
    const float* x     = (const float*)d_in[0];
    const float* W     = (const float*)d_in[1];
    const float* b     = (const float*)d_in[2];
    const float* gamma = (const float*)d_in[3];
    const float* beta  = (const float*)d_in[4];
    const float* ema   = (const float*)d_in[5];
    const int N    = in_sizes[2];              // 64
    const int D    = in_sizes[3];              // 2048
    const int rows = in_sizes[0] / D;          // B*T = 16384

    float* out        = (float*)d_out;
    float* out_sparse = out;                                   // rows*N
    float* out_idx    = out + (size_t)rows * N;                // rows*2 (as float)
    float* out_loss   = out_idx + (size_t)rows * 2;            // 1

    float* ws = (float*)d_ws;                  // [acc(64) | gdot(64) | bdot(64)]
    hipMemsetAsync(d_ws, 0, (size_t)N * sizeof(float), stream);

    precompute_dots<<<N, 256, 0, stream>>>(W, gamma, beta, ws);
    router_main<<<rows / 16, 128, 0, stream>>>(x, W, b, gamma,
                                               out_sparse, out_idx, ws);
    router_finalize<<<1, N, 0, stream>>>(ws, ema, out_loss, rows, N);
}